// EnhancedGNN_4569845202976
// MI455X (gfx1250) — compile-verified
//
#include <hip/hip_runtime.h>
#include <cmath>

#define N_NODES 100000
#define N_EDGES 1600000
#define HID 128
#define EPSV 1e-5f

typedef __attribute__((ext_vector_type(2))) float v2f;
typedef __attribute__((ext_vector_type(4))) float v4f;
typedef __attribute__((ext_vector_type(8))) float v8f;

// ---------------------------------------------------------------- utilities
__global__ void k_fill(float* __restrict__ p, float v, int n) {
    int i = blockIdx.x * blockDim.x + threadIdx.x;
    if (i < n) p[i] = v;
}

// deg[dst] += 1 per edge (deg pre-filled with 1.0 for the self loop)
__global__ void k_deg_accum(const int* __restrict__ dst, float* __restrict__ deg, int e) {
    int i = blockIdx.x * blockDim.x + threadIdx.x;
    if (i < e) unsafeAtomicAdd(&deg[dst[i]], 1.0f);
}

// in place: d = rsqrt(max(d,1))
__global__ void k_dinv(float* __restrict__ d, int n) {
    int i = blockIdx.x * blockDim.x + threadIdx.x;
    if (i < n) d[i] = rsqrtf(fmaxf(d[i], 1.0f));
}

// h = relu(x @ We + be), x:[n,2] We:[2,128]
__global__ void k_embed(const float* __restrict__ x, const float* __restrict__ We,
                        const float* __restrict__ be, float* __restrict__ h, int n) {
    int i = blockIdx.x * blockDim.x + threadIdx.x;
    if (i >= n * HID) return;
    int node = i >> 7, ch = i & (HID - 1);
    float v = fmaf(x[node * 2 + 0], We[ch],
              fmaf(x[node * 2 + 1], We[HID + ch], be[ch]));
    h[i] = fmaxf(v, 0.0f);
}

// Wt[c][k] = W[k][c]
__global__ void k_transpose(const float* __restrict__ W, float* __restrict__ Wt,
                            int K, int ncol) {
    int i = blockIdx.x * blockDim.x + threadIdx.x;
    if (i >= K * ncol) return;
    int k = i / ncol, c = i - k * ncol;
    Wt[(size_t)c * K + k] = W[i];
}

// ------------------------------------------------- fp32 WMMA GEMM  Out = A@W (+bias)
// A:[16*nRowTiles, K] row-major, Wt:[ncol, K] (W transposed).
// One wave -> 16 x (16*NT) output tile. All loads are base+immediate so the
// backend can clause-batch them and pipeline under the WMMAs.
// Optional fused epilogue: Agg = selfb + Out * dinv[row]^2 (GCN self-loop init).
template <int NT, int K>
__global__ __launch_bounds__(256) void k_gemm_wmma(
    const float* __restrict__ A, const float* __restrict__ Wt,
    const float* __restrict__ bias, float* __restrict__ Out,
    const float* __restrict__ dinv, const float* __restrict__ selfb,
    float* __restrict__ Agg, int nRowTiles, int ncol) {
    int colGroups = ncol / (16 * NT);
    int wave = (int)((blockIdx.x * blockDim.x + threadIdx.x) >> 5);
    int total = nRowTiles * colGroups;
    if (wave >= total) return;            // wave-uniform: EXEC stays all-ones
    int rt = wave / colGroups;
    int cg = wave - rt * colGroups;
    int ctBase = cg * NT;
    int lane = threadIdx.x & 31;
    int half = lane >> 4;                 // lanes 16..31 carry K+2 / K+3
    int l = lane & 15;

    const float* arow = A + (size_t)(rt * 16 + l) * K + 2 * half;      // row M=l
    // independent base pointer per column tile -> no dependent 64-bit addr chain
    const float* wp[NT];
#pragma unroll
    for (int t = 0; t < NT; t++)
        wp[t] = Wt + (size_t)((ctBase + t) * 16 + l) * K + 2 * half;   // col N=l

    v8f acc[NT] = {};
#pragma unroll 4
    for (int k0 = 0; k0 < K; k0 += 4) {
        // batch all loads (base + immediate), then the WMMA group
        v2f a = *(const v2f*)(arow + k0);
        v2f b[NT];
#pragma unroll
        for (int t = 0; t < NT; t++) b[t] = *(const v2f*)(wp[t] + k0);
#pragma unroll
        for (int t = 0; t < NT; t++)
            acc[t] = __builtin_amdgcn_wmma_f32_16x16x4_f32(
                false, a, false, b[t], (short)0, acc[t], false, false);
    }

    float di2[8];
    if (Agg) {
#pragma unroll
        for (int j = 0; j < 8; j++) {
            float d = dinv[rt * 16 + 8 * half + j];
            di2[j] = d * d;
        }
    }
#pragma unroll
    for (int t = 0; t < NT; t++) {
        int col = (ctBase + t) * 16 + l;
        float badd = bias ? bias[col] : 0.0f;
        size_t base = (size_t)(rt * 16 + 8 * half) * ncol + col;
#pragma unroll
        for (int j = 0; j < 8; j++) {                      // VGPR j: row j+8*half
            float v = acc[t][j] + badd;
            Out[base + (size_t)j * ncol] = v;
            if (Agg) Agg[base + (size_t)j * ncol] = fmaf(v, di2[j], selfb[col]);
        }
    }
}

// one wave per edge: agg[dst] += hw[src] * dinv[src]*dinv[dst] (128 ch = 32 lanes x float4)
__global__ __launch_bounds__(256) void k_scatter(
    const float* __restrict__ hw, const int* __restrict__ src, const int* __restrict__ dst,
    const float* __restrict__ dinv, float* __restrict__ agg, int e) {
    int wave = (int)((blockIdx.x * blockDim.x + threadIdx.x) >> 5);
    int lane = threadIdx.x & 31;
    if (wave >= e) return;
    int s = src[wave], d = dst[wave];
    float nrm = dinv[s] * dinv[d];
    v4f m = *((const v4f*)(hw + (size_t)s * HID) + lane);
    float* ap = agg + (size_t)d * HID + lane * 4;
    unsafeAtomicAdd(ap + 0, m.x * nrm);
    unsafeAtomicAdd(ap + 1, m.y * nrm);
    unsafeAtomicAdd(ap + 2, m.z * nrm);
    unsafeAtomicAdd(ap + 3, m.w * nrm);
}

// per-channel sum / sumsq over n rows; stats[0..c)=sum, stats[c..2c)=sumsq (pre-zeroed)
__global__ __launch_bounds__(256) void k_bn_stats(const float* __restrict__ v, int n, int c,
                                                  float* __restrict__ stats) {
    __shared__ float ls[256], lq[256];
    int tid = threadIdx.x;
    int ch = tid % c;
    int grp = tid / c;
    int rpb = 256 / c;
    float s = 0.0f, q = 0.0f;
    for (int r = blockIdx.x * rpb + grp; r < n; r += gridDim.x * rpb) {
        float x = v[(size_t)r * c + ch];
        s += x;
        q = fmaf(x, x, q);
    }
    ls[tid] = s; lq[tid] = q;
    __syncthreads();
    if (tid < c) {
        float ts = 0.0f, tq = 0.0f;
        for (int g = 0; g < rpb; g++) { ts += ls[g * c + tid]; tq += lq[g * c + tid]; }
        unsafeAtomicAdd(&stats[tid], ts);
        unsafeAtomicAdd(&stats[c + tid], tq);
    }
}

// in place: agg = relu(BN(agg)) + hres   (becomes next layer's h)
__global__ void k_bn_apply_res(float* __restrict__ agg, const float* __restrict__ hres,
                               const float* __restrict__ stats, const float* __restrict__ g,
                               const float* __restrict__ bt, int n) {
    int i = blockIdx.x * blockDim.x + threadIdx.x;
    if (i >= n * HID) return;
    int ch = i & (HID - 1);
    float invn = 1.0f / (float)n;
    float mu = stats[ch] * invn;
    float var = fmaf(-mu, mu, stats[HID + ch] * invn);
    float val = (agg[i] - mu) * rsqrtf(var + EPSV) * g[ch] + bt[ch];
    agg[i] = fmaxf(val, 0.0f) + hres[i];
}

// head: f = relu(BN(T)), out = tanh(f @ Wf2 + bf2); T:[n,32], Wf2:[32,2]
__global__ void k_final(const float* __restrict__ T, const float* __restrict__ stats,
                        const float* __restrict__ g, const float* __restrict__ bt,
                        const float* __restrict__ Wf2, const float* __restrict__ bf2,
                        float* __restrict__ out, int n) {
    int node = blockIdx.x * blockDim.x + threadIdx.x;
    if (node >= n) return;
    float f[32];
    float invn = 1.0f / (float)n;
#pragma unroll
    for (int c = 0; c < 32; c++) {
        float mu = stats[c] * invn;
        float var = fmaf(-mu, mu, stats[32 + c] * invn);
        float val = (T[(size_t)node * 32 + c] - mu) * rsqrtf(var + EPSV) * g[c] + bt[c];
        f[c] = fmaxf(val, 0.0f);
    }
#pragma unroll
    for (int j = 0; j < 2; j++) {
        float a = bf2[j];
#pragma unroll
        for (int c = 0; c < 32; c++) a = fmaf(f[c], Wf2[c * 2 + j], a);
        out[(size_t)node * 2 + j] = tanhf(a);
    }
}

// ---------------------------------------------------------------- launcher
extern "C" void kernel_launch(void* const* d_in, const int* in_sizes, int n_in,
                              void* d_out, int out_size, void* d_ws, size_t ws_size,
                              hipStream_t stream) {
    const int N = N_NODES, E = N_EDGES;
    const float* x   = (const float*)d_in[0];
    const int*   ei  = (const int*)d_in[1];           // [2, E]
    const int*   src = ei;
    const int*   dst = ei + E;
    const float* We  = (const float*)d_in[2];
    const float* be  = (const float*)d_in[3];
    const float* Wl[3]  = {(const float*)d_in[4],  (const float*)d_in[8],  (const float*)d_in[12]};
    const float* bl[3]  = {(const float*)d_in[5],  (const float*)d_in[9],  (const float*)d_in[13]};
    const float* gl[3]  = {(const float*)d_in[6],  (const float*)d_in[10], (const float*)d_in[14]};
    const float* btl[3] = {(const float*)d_in[7],  (const float*)d_in[11], (const float*)d_in[15]};
    const float* Wf1 = (const float*)d_in[16];
    const float* bf1 = (const float*)d_in[17];
    const float* gf  = (const float*)d_in[18];
    const float* btf = (const float*)d_in[19];
    const float* Wf2 = (const float*)d_in[20];
    const float* bf2 = (const float*)d_in[21];
    float* out = (float*)d_out;

    // workspace carve-up (everything fully initialized every call)
    char* w = (char*)d_ws;
    size_t off = 0;
    auto carve = [&](size_t bytes) -> float* {
        float* p = (float*)(w + off);
        off += (bytes + 255) & ~(size_t)255;
        return p;
    };
    const size_t big = (size_t)N * HID * sizeof(float);
    float* P0    = carve(big);
    float* P1    = carve(big);
    float* P2    = carve(big);
    float* T     = carve((size_t)N * 32 * sizeof(float));
    float* dinv  = carve((size_t)N * sizeof(float));
    float* stats = carve(256 * sizeof(float));
    float* Wt    = carve((size_t)HID * HID * sizeof(float));   // transposed weights
    (void)ws_size;

    const int TB = 256;
    const int nNH = N * HID;
    const int bNH = (nNH + TB - 1) / TB;
    const int bN  = (N + TB - 1) / TB;
    const int bE  = (E + TB - 1) / TB;
    const int bEw = (E * 32 + TB - 1) / TB;          // one wave per edge
    const int rowT = N / 16;                          // 6250 exactly

    // degrees -> dinv
    k_fill<<<bN, TB, 0, stream>>>(dinv, 1.0f, N);     // self loop
    k_deg_accum<<<bE, TB, 0, stream>>>(dst, dinv, E);
    k_dinv<<<bN, TB, 0, stream>>>(dinv, N);

    // embedding
    k_embed<<<bNH, TB, 0, stream>>>(x, We, be, P0, N);

    // three GCN blocks; buffers rotate (h, hw, agg) and BN-apply is in place in agg
    float* h   = P0;
    float* hw  = P1;
    float* agg = P2;
    const int bWT = (HID * HID + TB - 1) / TB;
    for (int layer = 0; layer < 3; layer++) {
        k_transpose<<<bWT, TB, 0, stream>>>(Wl[layer], Wt, HID, HID);
        // GEMM + fused self-loop/bias epilogue into agg
        int waves = rowT * (HID / 64);               // NT=4 -> 64 cols per wave
        int bG = (waves * 32 + TB - 1) / TB;
        k_gemm_wmma<4, HID><<<bG, TB, 0, stream>>>(h, Wt, nullptr, hw,
                                                   dinv, bl[layer], agg, rowT, HID);
        k_scatter<<<bEw, TB, 0, stream>>>(hw, src, dst, dinv, agg, E);
        k_fill<<<1, TB, 0, stream>>>(stats, 0.0f, 2 * HID);
        k_bn_stats<<<256, TB, 0, stream>>>(agg, N, HID, stats);
        k_bn_apply_res<<<bNH, TB, 0, stream>>>(agg, h, stats, gl[layer], btl[layer], N);
        // rotate: new h is agg; old h becomes next agg; hw slot is reused
        float* oldh = h;
        h = agg;
        agg = oldh;
    }

    // head: T = h3 @ Wf1 + bf1  (ncol = 32, NT=2 -> one col group)
    {
        const int bWT1 = (HID * 32 + TB - 1) / TB;
        k_transpose<<<bWT1, TB, 0, stream>>>(Wf1, Wt, HID, 32);
        int waves = rowT;
        int bG = (waves * 32 + TB - 1) / TB;
        k_gemm_wmma<2, HID><<<bG, TB, 0, stream>>>(h, Wt, bf1, T,
                                                   nullptr, nullptr, nullptr, rowT, 32);
    }
    k_fill<<<1, TB, 0, stream>>>(stats, 0.0f, 64);
    k_bn_stats<<<256, TB, 0, stream>>>(T, N, 32, stats);
    k_final<<<bN, TB, 0, stream>>>(T, stats, gf, btf, Wf2, bf2, out, N);
}